// GraphEncoder_43336220016663
// MI455X (gfx1250) — compile-verified
//
#include <hip/hip_runtime.h>
#include <hip/hip_bf16.h>
#include <cstddef>
#include <cstdint>

typedef __attribute__((ext_vector_type(2))) float v2f;
typedef __attribute__((ext_vector_type(8))) float v8f;

static inline unsigned cdiv_u(size_t a, unsigned b) { return (unsigned)((a + b - 1) / b); }

// ---------------------------------------------------------------------------
// Degree / normalization kernels
// ---------------------------------------------------------------------------
__global__ void deg_init(float* __restrict__ deg, int n) {
    int i = blockIdx.x * blockDim.x + threadIdx.x;
    if (i < n) deg[i] = 1.0f;                       // self-loop contributes 1.0
}

__global__ void deg_acc(const int* __restrict__ row, const float* __restrict__ val,
                        float* __restrict__ deg, int E) {
    int e = blockIdx.x * blockDim.x + threadIdx.x;
    if (e < E) atomicAdd(&deg[row[e]], val[e]);
}

__global__ void deg_rsqrt(float* __restrict__ deg, int n) {
    int i = blockIdx.x * blockDim.x + threadIdx.x;
    if (i < n) deg[i] = rsqrtf(deg[i]);             // deg >= 1, no zero guard needed
}

// ---------------------------------------------------------------------------
// Strip-mined dense GEMM via V_WMMA_F32_16X16X4_F32.
//   C[M x NOUT] = A[M x K] @ W[K x NOUT] (+bias)(+relu),  NOUT = NT*16
// One wave computes a full 16 x NOUT row strip: each A fragment (b64 global
// load) feeds NT WMMAs on independent accumulators. W is staged in LDS in
// KC-row chunks, PRE-PAIRED along K: element (k,n) lives in float2 slot
// [k/2][n], so a lane's B fragment (W[k+kh][n], W[k+kh+1][n]) is one
// contiguous ds_load_b64 straight into an aligned VGPR pair (no shuffles).
// NPITCH = NOUT+16 (≡16 mod 32) rotates banks by 32 between the two
// half-wave K-groups -> conflict-free b64 reads.
// Requires M%16==0 and K%KC==0 (K in {64,128,256}).
// ---------------------------------------------------------------------------
template <int NT>
__global__ void gemm_wmma_strip(const float* __restrict__ A,
                                const float* __restrict__ W,
                                const float* __restrict__ bias,
                                float* __restrict__ C,
                                int M, int K, int relu) {
    constexpr int KC     = 64;              // K rows staged per chunk
    constexpr int NOUT   = NT * 16;
    constexpr int NPITCH = NOUT + 16;       // float2 pitch per K-pair row
    __shared__ float Ws[(KC / 2) * NPITCH * 2];   // (KC/2) x NPITCH float2 slots

    const int tid    = threadIdx.x;
    const int lane   = tid & 31;
    const int wave   = tid >> 5;
    const int l15    = lane & 15;
    const int khalf  = (lane >> 4) << 1;    // lanes 0-15 -> K{0,1}, 16-31 -> K{2,3}
    const int kpairH = lane >> 4;           // pair offset 0 or 1

    const int tilesM = M >> 4;
    int tm = blockIdx.x * (blockDim.x >> 5) + wave;
    const bool active = tm < tilesM;
    if (!active) tm = tilesM - 1;           // clamp: loads stay in-bounds

    const float* Arow = A + (size_t)((tm << 4) + l15) * K + khalf;

    v8f acc[NT];
#pragma unroll
    for (int t = 0; t < NT; ++t) acc[t] = (v8f){0.f,0.f,0.f,0.f,0.f,0.f,0.f,0.f};

    for (int kc = 0; kc < K; kc += KC) {
        __syncthreads();                    // previous chunk fully consumed
        // Cooperative stage: global W[kc+kk][nn] -> Ws pair slot [kk>>1][nn].(kk&1)
        for (int i = tid; i < KC * NOUT; i += blockDim.x) {
            const int kk = i >> (NT == 8 ? 7 : 6);      // i / NOUT
            const int nn = i & (NOUT - 1);
            Ws[(size_t)(kk >> 1) * (NPITCH * 2) + nn * 2 + (kk & 1)] =
                W[(size_t)(kc + kk) * NOUT + nn];
        }
        __syncthreads();

#pragma unroll 4
        for (int k = 0; k < KC; k += 4) {
            v2f a;
            a.x = Arow[kc + k];
            a.y = Arow[kc + k + 1];
            const int prow = ((k >> 1) + kpairH) * (NPITCH * 2);
#pragma unroll
            for (int t = 0; t < NT; ++t) {
                const v2f b = *(const v2f*)&Ws[prow + (t * 16 + l15) * 2];
                acc[t] = __builtin_amdgcn_wmma_f32_16x16x4_f32(
                    /*neg_a=*/false, a, /*neg_b=*/false, b,
                    /*c_mod=*/(short)0, acc[t], /*reuse_a=*/false, /*reuse_b=*/false);
            }
        }
    }

    if (!active) return;                    // wave-uniform
    const int rbase = (tm << 4) + ((lane >> 4) << 3);   // C: M = r + 8*(lane>>4)
#pragma unroll
    for (int t = 0; t < NT; ++t) {
        const int ncol = t * 16 + l15;
        const float bv = bias ? bias[ncol] : 0.0f;
#pragma unroll
        for (int r = 0; r < 8; ++r) {
            float v = acc[t][r] + bv;
            if (relu) v = fmaxf(v, 0.0f);
            C[(size_t)(rbase + r) * NOUT + ncol] = v;
        }
    }
}

// ---------------------------------------------------------------------------
// SpMM pieces: out[row] += dinv[row]*val*dinv[col] * P[col], plus self-loop.
// ---------------------------------------------------------------------------
__global__ void spmm_self(const float* __restrict__ P, const float* __restrict__ dinv,
                          float* __restrict__ out, int n, int logF) {
    const size_t i = (size_t)blockIdx.x * blockDim.x + threadIdx.x;   // float4 index
    const size_t total = ((size_t)n << logF) >> 2;
    if (i >= total) return;
    const int node = (int)(i >> (logF - 2));
    float d = dinv[node];
    d = d * d;                                       // self-loop weight dinv_i * 1 * dinv_i
    const float4 p = ((const float4*)P)[i];
    float4 o;
    o.x = d * p.x; o.y = d * p.y; o.z = d * p.z; o.w = d * p.w;
    ((float4*)out)[i] = o;
}

__global__ void spmm_edges(const int* __restrict__ row, const int* __restrict__ col,
                           const float* __restrict__ val, const float* __restrict__ dinv,
                           const float* __restrict__ P, float* __restrict__ out,
                           int E, int F, int logPer) {
    const size_t tid = (size_t)blockIdx.x * blockDim.x + threadIdx.x;
    const size_t total = (size_t)E << logPer;        // per = F/4 threads per edge
    if (tid >= total) return;
    const int e = (int)(tid >> logPer);
    const int f = (int)(tid & ((1u << logPer) - 1)) << 2;
    const int r = row[e], c = col[e];
    const float v = dinv[r] * val[e] * dinv[c];
    const float4 h = *(const float4*)(P + (size_t)c * F + f);
    float* o = out + (size_t)r * F + f;
    atomicAdd(o + 0, v * h.x);
    atomicAdd(o + 1, v * h.y);
    atomicAdd(o + 2, v * h.z);
    atomicAdd(o + 3, v * h.w);
}

__global__ void bias_act(float* __restrict__ out, const float* __restrict__ bias,
                         int n, int logF, int relu) {
    const size_t i = (size_t)blockIdx.x * blockDim.x + threadIdx.x;   // float4 index
    const size_t total = ((size_t)n << logF) >> 2;
    if (i >= total) return;
    const int fvec = (int)(i & ((1u << (logF - 2)) - 1));
    const float4 b = ((const float4*)bias)[fvec];
    float4 v = ((float4*)out)[i];
    v.x += b.x; v.y += b.y; v.z += b.z; v.w += b.w;
    if (relu) {
        v.x = fmaxf(v.x, 0.f); v.y = fmaxf(v.y, 0.f);
        v.z = fmaxf(v.z, 0.f); v.w = fmaxf(v.w, 0.f);
    }
    ((float4*)out)[i] = v;
}

// ---------------------------------------------------------------------------
extern "C" void kernel_launch(void* const* d_in, const int* in_sizes, int n_in,
                              void* d_out, int out_size, void* d_ws, size_t ws_size,
                              hipStream_t stream) {
    const float* x   = (const float*)d_in[0];
    const int*   row = (const int*)  d_in[1];
    const int*   col = (const int*)  d_in[2];
    const float* val = (const float*)d_in[3];
    // d_in[4] = training flag (eval mode: identity dropout) — ignored
    const float* W1  = (const float*)d_in[5];
    const float* b1  = (const float*)d_in[6];
    const float* W2  = (const float*)d_in[7];
    const float* b2  = (const float*)d_in[8];
    const float* W3  = (const float*)d_in[9];
    const float* b3  = (const float*)d_in[10];
    const float* Wp1 = (const float*)d_in[11];
    const float* bp1 = (const float*)d_in[12];
    const float* Wp2 = (const float*)d_in[13];
    const float* bp2 = (const float*)d_in[14];

    const int IN = 256, HID = 128, EMB = 64, PROJ = 64;
    const int N = in_sizes[0] / IN;                  // 50000 (divisible by 16)
    const int E = in_sizes[1];                       // 800000

    float* x_emb = (float*)d_out;                    // N x EMB
    float* z     = (float*)d_out + (size_t)N * EMB;  // N x PROJ

    float* dinv = (float*)d_ws;                      // N floats (N*4 % 16 == 0)
    float* buf0 = dinv + N;                          // N x HID
    float* buf1 = buf0 + (size_t)N * HID;            // N x HID

    const unsigned B = 256;

    // --- gcn_norm: dinv = rsqrt(1 + segment_sum(val, row)) ---
    deg_init <<<cdiv_u(N, B), B, 0, stream>>>(dinv, N);
    deg_acc  <<<cdiv_u(E, B), B, 0, stream>>>(row, val, dinv, E);
    deg_rsqrt<<<cdiv_u(N, B), B, 0, stream>>>(dinv, N);

    const int tilesM = N >> 4;                       // 3125 = 5^5
    const unsigned WV = 5;                           // waves per block (3125 % 5 == 0)
    const unsigned gemmGrid  = cdiv_u(tilesM, WV);   // 625 blocks, no idle waves
    const unsigned gemmBlock = WV * 32;              // 160 threads

    // --- Layer 1: buf1 = relu(spmm(x @ W1) + b1) ---
    gemm_wmma_strip<8><<<gemmGrid, gemmBlock, 0, stream>>>(x, W1, nullptr, buf0, N, IN, 0);
    spmm_self <<<cdiv_u(((size_t)N * HID) >> 2, B), B, 0, stream>>>(buf0, dinv, buf1, N, 7);
    spmm_edges<<<cdiv_u((size_t)E << 5, B), B, 0, stream>>>(row, col, val, dinv, buf0, buf1, E, HID, 5);
    bias_act  <<<cdiv_u(((size_t)N * HID) >> 2, B), B, 0, stream>>>(buf1, b1, N, 7, 1);

    // --- Layer 2: buf1 = relu(spmm(buf1 @ W2) + b2) ---
    gemm_wmma_strip<8><<<gemmGrid, gemmBlock, 0, stream>>>(buf1, W2, nullptr, buf0, N, HID, 0);
    spmm_self <<<cdiv_u(((size_t)N * HID) >> 2, B), B, 0, stream>>>(buf0, dinv, buf1, N, 7);
    spmm_edges<<<cdiv_u((size_t)E << 5, B), B, 0, stream>>>(row, col, val, dinv, buf0, buf1, E, HID, 5);
    bias_act  <<<cdiv_u(((size_t)N * HID) >> 2, B), B, 0, stream>>>(buf1, b2, N, 7, 1);

    // --- Layer 3: x_emb = spmm(buf1 @ W3) + b3  (no relu) ---
    gemm_wmma_strip<4><<<gemmGrid, gemmBlock, 0, stream>>>(buf1, W3, nullptr, buf0, N, HID, 0);
    spmm_self <<<cdiv_u(((size_t)N * EMB) >> 2, B), B, 0, stream>>>(buf0, dinv, x_emb, N, 6);
    spmm_edges<<<cdiv_u((size_t)E << 4, B), B, 0, stream>>>(row, col, val, dinv, buf0, x_emb, E, EMB, 4);
    bias_act  <<<cdiv_u(((size_t)N * EMB) >> 2, B), B, 0, stream>>>(x_emb, b3, N, 6, 0);

    // --- Projection head: z = relu(x_emb @ Wp1 + bp1) @ Wp2 + bp2 ---
    gemm_wmma_strip<4><<<gemmGrid, gemmBlock, 0, stream>>>(x_emb, Wp1, bp1, buf0, N, EMB, 1);
    gemm_wmma_strip<4><<<gemmGrid, gemmBlock, 0, stream>>>(buf0, Wp2, bp2, z, N, PROJ, 0);
}